// QuantumConvLayer3_7980049236671
// MI455X (gfx1250) — compile-verified
//
#include <hip/hip_runtime.h>
#include <math.h>

#ifndef __has_builtin
#define __has_builtin(x) 0
#endif

// CDNA5 async global->LDS data mover path (ASYNCcnt-tracked), with a safe
// fallback to plain b128 loads if the builtin isn't declared by this clang.
#if defined(__HIP_DEVICE_COMPILE__) && __has_builtin(__builtin_amdgcn_global_load_async_to_lds_b128)
#define QC_ASYNC_LDS 1
#else
#define QC_ASYNC_LDS 0
#endif

#if QC_ASYNC_LDS
typedef int qc_v4i __attribute__((ext_vector_type(4)));
typedef qc_v4i __attribute__((address_space(1))) * qc_gvp;   // global v4i*
typedef qc_v4i __attribute__((address_space(3))) * qc_lvp;   // LDS v4i*
#endif

// cos(p0 + th0) for p0 in [0,pi], th0 in [0,2pi]  =>  arg in [0,3pi].
// Use v_cos_f32 (input in revolutions, hardware does fractional reduction).
// rev = fma(p0, 1/2pi, th0/2pi); phase rounding error <= ~4e-7 rad, same
// order as the reference's own fp32 accumulation error.
__device__ __forceinline__ float qc_cos_sum(float p0, float th0_rev) {
#if defined(__HIP_DEVICE_COMPILE__) && __has_builtin(__builtin_amdgcn_cosf)
    const float inv2pi = 0.15915494309189535f;
    return __builtin_amdgcn_cosf(fmaf(p0, inv2pi, th0_rev));
#else
    const float twopi = 6.283185307179586f;
    return cosf(fmaf(th0_rev, twopi, p0));
#endif
}

// out[b] = cos(patches[b,0] + theta[0])   (exact closed form of the circuit:
// RY(theta)·RY(p) = RY(p+theta), and both CNOTs preserve the qubit-0 marginal)
__global__ __launch_bounds__(256) void qconv3_z0_kernel(
    const float* __restrict__ patches,
    const float* __restrict__ theta,
    float* __restrict__ out,
    int npatch)
{
    const int nquads = npatch >> 2;          // 4 patches per thread
    const int tid    = threadIdx.x;
    const int t      = blockIdx.x * 256 + tid;
    const float th0_rev = theta[0] * 0.15915494309189535f;  // uniform scalar

#if QC_ASYNC_LDS
    __shared__ float tile[12 * 256];         // 48 B per thread = 12 KB

    const bool have = (t < nquads);
    if (have) {
        qc_gvp g = (qc_gvp)(patches + (size_t)t * 12);
        qc_lvp l = (qc_lvp)&tile[tid * 12];
        // three async 16B copies: global -> LDS, bypassing VGPRs
        __builtin_amdgcn_global_load_async_to_lds_b128(g,     l,     0, 0);
        __builtin_amdgcn_global_load_async_to_lds_b128(g + 1, l + 1, 0, 0);
        __builtin_amdgcn_global_load_async_to_lds_b128(g + 2, l + 2, 0, 0);
    }
#if __has_builtin(__builtin_amdgcn_s_wait_asynccnt)
    __builtin_amdgcn_s_wait_asynccnt(0);
#else
    asm volatile("s_wait_asynccnt 0x0" ::: "memory");
#endif
    if (have) {
        // this wave staged its own 48B region; the asynccnt wait is sufficient
        const float* m = &tile[tid * 12];
        float4 r;
        r.x = qc_cos_sum(m[0], th0_rev);   // p0 of patch 4t+0
        r.y = qc_cos_sum(m[3], th0_rev);   // p0 of patch 4t+1
        r.z = qc_cos_sum(m[6], th0_rev);   // p0 of patch 4t+2
        r.w = qc_cos_sum(m[9], th0_rev);   // p0 of patch 4t+3
        reinterpret_cast<float4*>(out)[t] = r;
    }
#else
    if (t < nquads) {
        const float4* p4 = reinterpret_cast<const float4*>(patches) + (size_t)t * 3;
        float4 a = p4[0];
        float4 b = p4[1];
        float4 c = p4[2];
        float4 r;
        r.x = qc_cos_sum(a.x, th0_rev);    // float index 12t + 0
        r.y = qc_cos_sum(a.w, th0_rev);    // 12t + 3
        r.z = qc_cos_sum(b.z, th0_rev);    // 12t + 6
        r.w = qc_cos_sum(c.y, th0_rev);    // 12t + 9
        reinterpret_cast<float4*>(out)[t] = r;
    }
#endif

    // tail patches (npatch % 4) handled scalarly by block 0
    const int tail = npatch & 3;
    if (blockIdx.x == 0 && tid < tail) {
        const int p = nquads * 4 + tid;
        out[p] = qc_cos_sum(patches[(size_t)p * 3], th0_rev);
    }
}

extern "C" void kernel_launch(void* const* d_in, const int* in_sizes, int n_in,
                              void* d_out, int out_size, void* d_ws, size_t ws_size,
                              hipStream_t stream) {
    const float* patches = (const float*)d_in[0];   // [B,3] f32
    const float* theta   = (const float*)d_in[1];   // [3]   f32
    float* out = (float*)d_out;                     // [B]   f32

    const int npatch = out_size;                    // B = 2,097,152
    const int nquads = npatch / 4;
    int blocks = (nquads + 255) / 256;
    if (blocks < 1) blocks = 1;

    qconv3_z0_kernel<<<blocks, 256, 0, stream>>>(patches, theta, out, npatch);
}